// Selector_67525475828317
// MI455X (gfx1250) — compile-verified
//
#include <hip/hip_runtime.h>
#include <hip/hip_bf16.h>

typedef __attribute__((ext_vector_type(2))) float v2f;
typedef __attribute__((ext_vector_type(8))) float v8f;

#define KC 128                 // K-chunk staged in LDS
#define RP 64                  // relations padded to 4 WMMA column tiles
#define WAVES_PER_BLOCK 8
#define BLOCK (WAVES_PER_BLOCK * 32)
#define KSTEPS (KC / 4)        // WMMA k-steps per chunk
#define FRAGS (KSTEPS * 4 * 32)  // float2 fragment slots per chunk (32 KB)

// MODE 0: score[i] = sum_r softmax(x@W+b)[i,r] * knowledge[i,r]   (rows = N instances)
// MODE 1: out[b,:] = softmax(x[best[b]]@W+b)                      (rows = B bags)
template <int MODE>
__global__ __launch_bounds__(BLOCK) void gemm_softmax_kernel(
    const float* __restrict__ x, const float* __restrict__ knowledge,
    const float* __restrict__ rel, const float* __restrict__ bias,
    const int* __restrict__ best, float* __restrict__ outv,
    int nRows, int H, int R, int N) {
  // B fragments pre-swizzled to WMMA layout: [ks][tile][lane] -> {rel[kk][col], rel[kk+1][col]}
  __shared__ v2f ldsB[FRAGS];  // 32 KB

  const int tid = threadIdx.x;
  const int lane = tid & 31;
  const int wave = tid >> 5;
  const int nlane = lane & 15;
  const int half = lane >> 4;

  const int numTiles = (nRows + 15) >> 4;
  int tile = blockIdx.x * WAVES_PER_BLOCK + wave;
  if (tile >= numTiles) tile = numTiles - 1;  // keep all waves alive for barriers
  const int rowBase = tile << 4;

  // WMMA 16x4 f32 A layout: M = lane&15 ; VGPR k-pair offset = 2*(lane>>4)
  int aRow = rowBase + nlane;
  if (aRow > nRows - 1) aRow = nRows - 1;
  long aIdx;
  if (MODE == 1) {
    int bi = best[aRow];
    if (bi < 0) bi = 0;
    if (bi > N - 1) bi = N - 1;
    aIdx = (long)bi * H;
  } else {
    aIdx = (long)aRow * H;
  }
  const int kOff = half * 2;

  v8f acc0 = {}, acc1 = {}, acc2 = {}, acc3 = {};

  for (int kc = 0; kc < H; kc += KC) {
    __syncthreads();
    // Stage rel[kc:kc+KC, 0:R] (zero-padded cols) directly in B-fragment order.
    for (int s = tid; s < FRAGS; s += BLOCK) {
      const int ks = s >> 7;        // k-step within chunk
      const int t = (s >> 5) & 3;   // column tile
      const int l = s & 31;         // lane slot
      const int col = t * 16 + (l & 15);
      const int kk = kc + ks * 4 + (l >> 4) * 2;
      v2f p;
      if (col < R) {
        p.x = rel[(long)kk * R + col];
        p.y = rel[(long)(kk + 1) * R + col];
      } else {
        p.x = 0.f;
        p.y = 0.f;
      }
      ldsB[s] = p;
    }
    __syncthreads();

    for (int ks = 0; ks < KSTEPS; ++ks) {
      const float* ap = x + aIdx + (kc + ks * 4 + kOff);
      v2f a = *(const v2f*)ap;  // 8-byte aligned (even k)
      const int fb = ks * 128 + lane;  // one base reg + imm offsets (<=32 KB)
      v2f b0 = ldsB[fb + 0];
      v2f b1 = ldsB[fb + 32];
      v2f b2 = ldsB[fb + 64];
      v2f b3 = ldsB[fb + 96];
      acc0 = __builtin_amdgcn_wmma_f32_16x16x4_f32(false, a, false, b0, (short)0, acc0, false, false);
      acc1 = __builtin_amdgcn_wmma_f32_16x16x4_f32(false, a, false, b1, (short)0, acc1, false, false);
      acc2 = __builtin_amdgcn_wmma_f32_16x16x4_f32(false, a, false, b2, (short)0, acc2, false, false);
      acc3 = __builtin_amdgcn_wmma_f32_16x16x4_f32(false, a, false, b3, (short)0, acc3, false, false);
    }
  }

  // ---- epilogue: C layout puts col n=nlane in each lane; rows 0..7 in lanes
  // 0-15 (vgpr j), rows 8..15 in lanes 16-31 ----
  const int col0 = nlane, col1 = nlane + 16, col2 = nlane + 32, col3 = nlane + 48;
  const bool v0 = col0 < R, v1 = col1 < R, v2 = col2 < R, v3 = col3 < R;
  const float bb0 = v0 ? bias[col0] : 0.f;
  const float bb1 = v1 ? bias[col1] : 0.f;
  const float bb2 = v2 ? bias[col2] : 0.f;
  const float bb3 = v3 ? bias[col3] : 0.f;

#pragma unroll
  for (int j = 0; j < 8; ++j) {
    const int row = rowBase + half * 8 + j;
    float l0 = acc0[j] + bb0;
    float l1 = acc1[j] + bb1;
    float l2 = acc2[j] + bb2;
    float l3 = acc3[j] + bb3;

    float mx = -__builtin_inff();
    if (v0) mx = fmaxf(mx, l0);
    if (v1) mx = fmaxf(mx, l1);
    if (v2) mx = fmaxf(mx, l2);
    if (v3) mx = fmaxf(mx, l3);
#pragma unroll
    for (int m = 1; m < 16; m <<= 1) mx = fmaxf(mx, __shfl_xor(mx, m, 32));

    float e0 = v0 ? __expf(l0 - mx) : 0.f;
    float e1 = v1 ? __expf(l1 - mx) : 0.f;
    float e2 = v2 ? __expf(l2 - mx) : 0.f;
    float e3 = v3 ? __expf(l3 - mx) : 0.f;
    float se = e0 + e1 + e2 + e3;
#pragma unroll
    for (int m = 1; m < 16; m <<= 1) se += __shfl_xor(se, m, 32);

    if (MODE == 0) {
      const int rowc = row < N ? row : N - 1;
      const long kb = (long)rowc * R;
      float sk = 0.f;
      if (v0) sk += e0 * knowledge[kb + col0];
      if (v1) sk += e1 * knowledge[kb + col1];
      if (v2) sk += e2 * knowledge[kb + col2];
      if (v3) sk += e3 * knowledge[kb + col3];
#pragma unroll
      for (int m = 1; m < 16; m <<= 1) sk += __shfl_xor(sk, m, 32);
      if (nlane == 0 && row < nRows) outv[row] = sk / se;
    } else {
      if (row < nRows) {
        const long ob = (long)row * R;
        const float inv = 1.0f / se;
        if (v0) outv[ob + col0] = e0 * inv;
        if (v1) outv[ob + col1] = e1 * inv;
        if (v2) outv[ob + col2] = e2 * inv;
        if (v3) outv[ob + col3] = e3 * inv;
      }
    }
  }
}

// First-index-of-max within each contiguous bag (matches segment_max + min-index).
__global__ void argmax_bag_kernel(const float* __restrict__ score,
                                  const int* __restrict__ scope,
                                  int* __restrict__ best, int B) {
  int b = blockIdx.x * blockDim.x + threadIdx.x;
  if (b >= B) return;
  int s = scope[2 * b];
  int e = scope[2 * b + 1];
  float bm = -__builtin_inff();
  int bi = s;
  for (int i = s; i < e; ++i) {
    float sc = score[i];
    if (sc > bm) { bm = sc; bi = i; }
  }
  best[b] = bi;
}

__global__ void copy_kernel(const float* __restrict__ src, float* __restrict__ dst, int n) {
  int i = blockIdx.x * blockDim.x + threadIdx.x;
  if (i < n) dst[i] = src[i];
}

extern "C" void kernel_launch(void* const* d_in, const int* in_sizes, int n_in,
                              void* d_out, int out_size, void* d_ws, size_t ws_size,
                              hipStream_t stream) {
  const float* x = (const float*)d_in[0];
  const float* knowledge = (const float*)d_in[1];
  const float* rel = (const float*)d_in[2];
  const float* bias = (const float*)d_in[3];
  const int* scope = (const int*)d_in[4];

  const int R = in_sizes[3];      // 53
  const int H = in_sizes[2] / R;  // 768
  const int N = in_sizes[0] / H;  // 100000
  const int B = in_sizes[4] / 2;  // 20000

  float* out = (float*)d_out;  // [B*R bag_probs][H*R rel_mat]
  float* score = (float*)d_ws; // N floats
  int* best = (int*)((char*)d_ws + (((size_t)N * sizeof(float) + 255) & ~(size_t)255));

  // 1) per-instance scores via fp32 WMMA GEMM + fused softmax/weighted-sum
  {
    int tiles = (N + 15) / 16;
    int blocks = (tiles + WAVES_PER_BLOCK - 1) / WAVES_PER_BLOCK;
    gemm_softmax_kernel<0><<<blocks, BLOCK, 0, stream>>>(
        x, knowledge, rel, bias, nullptr, score, N, H, R, N);
  }
  // 2) per-bag first-argmax over contiguous ranges
  argmax_bag_kernel<<<(B + 255) / 256, 256, 0, stream>>>(score, scope, best, B);
  // 3) bag probs via same WMMA kernel with row indirection
  {
    int tiles = (B + 15) / 16;
    int blocks = (tiles + WAVES_PER_BLOCK - 1) / WAVES_PER_BLOCK;
    gemm_softmax_kernel<1><<<blocks, BLOCK, 0, stream>>>(
        x, nullptr, rel, bias, best, out, B, H, R, N);
  }
  // 4) second tuple element: rel_mat passthrough
  {
    int n = H * R;
    copy_kernel<<<(n + 255) / 256, 256, 0, stream>>>(rel, out + (size_t)B * R, n);
  }
}